// Conv_1563368096281
// MI455X (gfx1250) — compile-verified
//
#include <hip/hip_runtime.h>
#include <stdint.h>

// ---------------------------------------------------------------------------
// Problem constants (from reference): B=2, Np=256, Ny=512, No=256, C=64
// ---------------------------------------------------------------------------
constexpr int BB  = 2;
constexpr int NP  = 256;
constexpr int NY  = 512;
constexpr int NO  = 256;
constexpr int N   = NP + NY + NO;   // 1024
constexpr int C   = 64;
constexpr int C2  = 128;            // 2*C
constexpr int TI  = 16;             // WMMA tile rows (M)
constexpr int NWAVE = 4;            // waves per block in main kernel
constexpr int NJT  = (N / TI) / NWAVE;  // j-tiles per wave = 16

typedef __attribute__((ext_vector_type(2))) float v2f;
typedef __attribute__((ext_vector_type(8))) float v8f;

__device__ __forceinline__ float eluf(float x) {
    return x > 0.0f ? x : (__expf(x) - 1.0f);
}

// Output tuple (pred, prey, obst) concatenated flat in return order.
__device__ __forceinline__ size_t out_off(int b, int n) {
    if (n < NP)      return ((size_t)b * NP + n) * C;
    if (n < NP + NY) return (size_t)BB * NP * C + ((size_t)b * NY + (n - NP)) * C;
    return (size_t)BB * NP * C + (size_t)BB * NY * C + ((size_t)b * NO + (n - NP - NY)) * C;
}

// ---------------------------------------------------------------------------
// Kernel 1: build h = concat(x_pred, x_prey, x_obst) and u = h @ conv1_w^T
// ---------------------------------------------------------------------------
__global__ void __launch_bounds__(128) prep_kernel(
    const float* __restrict__ xp, const float* __restrict__ xy,
    const float* __restrict__ xo, const float* __restrict__ w1,
    float* __restrict__ h, float* __restrict__ u) {
    __shared__ __align__(16) float row[C];
    int bn = blockIdx.x;
    int b  = bn / N;
    int n  = bn % N;
    const float* src;
    if (n < NP)           src = xp + ((size_t)b * NP + n) * C;
    else if (n < NP + NY) src = xy + ((size_t)b * NY + (n - NP)) * C;
    else                  src = xo + ((size_t)b * NO + (n - NP - NY)) * C;

    int t = threadIdx.x;
    if (t < C) {
        float v = src[t];
        row[t] = v;
        h[(size_t)bn * C + t] = v;
    }
    __syncthreads();

    const float* wr = w1 + (size_t)t * C;   // conv1_w[o, :]
    float acc = 0.0f;
#pragma unroll
    for (int c = 0; c < C; ++c) acc = fmaf(row[c], wr[c], acc);
    u[(size_t)bn * C2 + t] = acc;
}

// ---------------------------------------------------------------------------
// Kernel 2 (main, fused): per 16-row i-tile:
//   j-loop: async-copy u_j tile to LDS (GLOBAL_LOAD_ASYNC_TO_LDS_B128),
//           compute W subtile (elu-dot, VALU) in WMMA A-fragment layout,
//           agg += W @ h via v_wmma_f32_16x16x4_f32 (B-fragments from L2);
//   epilogue: out = h + elu(agg @ lin_w^T + lin_b)
// block = 128 threads (4 wave32); grid = B * N/16 = 128
// ---------------------------------------------------------------------------
__global__ void __launch_bounds__(128) edge_agg_kernel(
    const float* __restrict__ h,  const float* __restrict__ u,
    const float* __restrict__ b1, const float* __restrict__ w2,
    const float* __restrict__ linw, const float* __restrict__ linb,
    const float* __restrict__ c2b, float* __restrict__ out) {

    __shared__ __align__(16) float uI[TI * C2];             // 8 KB, u_i + b1 folded
    __shared__ __align__(16) float uJbuf[NWAVE * TI * C2];  // 32 KB; reused as agg partials
    __shared__ __align__(16) float w2s[C2];

    const int tid  = threadIdx.x;
    const int wave = tid >> 5;
    const int lane = tid & 31;
    const int row  = lane & 15;       // M within tile
    const int half = lane >> 4;       // 0 or 1
    const int ncol = lane & 15;       // N within C-block

    const int blk   = blockIdx.x;
    const int b     = blk / (N / TI);
    const int iTile = blk % (N / TI);
    const int iBase = iTile * TI;

    const float* ub = u + (size_t)b * N * C2;
    const float* hb = h + (size_t)b * N * C;

    // load i-tile of u with conv1_b pre-added (z = (u_i + b1) - u_j)
    for (int idx = tid * 4; idx < TI * C2; idx += 128 * 4) {
        float4 a  = *(const float4*)&ub[(size_t)iBase * C2 + idx];
        float4 bb = *(const float4*)&b1[idx & (C2 - 1)];
        float4 r;
        r.x = a.x + bb.x; r.y = a.y + bb.y; r.z = a.z + bb.z; r.w = a.w + bb.w;
        *(float4*)&uI[idx] = r;
    }
    if (tid < C2) w2s[tid] = w2[tid];
    __syncthreads();

    const float c2bias = c2b[0];
    float* uJw = &uJbuf[wave * TI * C2];

    v8f accs[4];
#pragma unroll
    for (int cb = 0; cb < 4; ++cb) accs[cb] = (v8f){};

    // uniform trip count: no EXEC-masked looping around the WMMA body
    for (int it = 0; it < NJT; ++it) {
        const int jt    = it * NWAVE + wave;   // wave-uniform
        const int jBase = jt * TI;

        if (it + 1 < NJT)   // prefetch next j-tile of u into L2
            __builtin_prefetch(&ub[(size_t)(jBase + NWAVE * TI) * C2], 0, 0);

        // ---- async copy u_j tile (16x128 f32 = 8 KB) global -> LDS ----
        // per lane: 16B at lane*16 + i*512, i = 0..15 (same inst offset both sides)
        {
            unsigned la = (unsigned)(uintptr_t)(void*)(uJw + lane * 4);
            unsigned long long ga =
                (unsigned long long)(uintptr_t)(ub + (size_t)jBase * C2 + lane * 4);
            asm volatile(
                "global_load_async_to_lds_b128 %0, %1, off\n\t"
                "global_load_async_to_lds_b128 %0, %1, off offset:512\n\t"
                "global_load_async_to_lds_b128 %0, %1, off offset:1024\n\t"
                "global_load_async_to_lds_b128 %0, %1, off offset:1536\n\t"
                "global_load_async_to_lds_b128 %0, %1, off offset:2048\n\t"
                "global_load_async_to_lds_b128 %0, %1, off offset:2560\n\t"
                "global_load_async_to_lds_b128 %0, %1, off offset:3072\n\t"
                "global_load_async_to_lds_b128 %0, %1, off offset:3584\n\t"
                "global_load_async_to_lds_b128 %0, %1, off offset:4096\n\t"
                "global_load_async_to_lds_b128 %0, %1, off offset:4608\n\t"
                "global_load_async_to_lds_b128 %0, %1, off offset:5120\n\t"
                "global_load_async_to_lds_b128 %0, %1, off offset:5632\n\t"
                "global_load_async_to_lds_b128 %0, %1, off offset:6144\n\t"
                "global_load_async_to_lds_b128 %0, %1, off offset:6656\n\t"
                "global_load_async_to_lds_b128 %0, %1, off offset:7168\n\t"
                "global_load_async_to_lds_b128 %0, %1, off offset:7680"
                :: "v"(la), "v"(ga) : "memory");
            asm volatile("s_wait_asynccnt 0x0" ::: "memory");
        }

        // ---- compute 16x16 W subtile in WMMA A-fragment layout ----
        // lane holds W[row][4q + 2*half + t], q=0..3, t=0..1
        float wv[8];
#pragma unroll
        for (int e = 0; e < 8; ++e) wv[e] = 0.0f;

        const int c0 = 2 * half;
        for (int k = 0; k < C2; k += 4) {
            float4 ti4 = *(const float4*)&uI[row * C2 + k];   // u_i + b1
            float4 wwv = *(const float4*)&w2s[k];
#pragma unroll
            for (int q = 0; q < 4; ++q) {
#pragma unroll
                for (int t2 = 0; t2 < 2; ++t2) {
                    const int col = 4 * q + c0 + t2;
                    float4 uj = *(const float4*)&uJw[col * C2 + k];
                    float a = wv[q * 2 + t2];
                    a = fmaf(eluf(ti4.x - uj.x), wwv.x, a);
                    a = fmaf(eluf(ti4.y - uj.y), wwv.y, a);
                    a = fmaf(eluf(ti4.z - uj.z), wwv.z, a);
                    a = fmaf(eluf(ti4.w - uj.w), wwv.w, a);
                    wv[q * 2 + t2] = a;
                }
            }
        }
#pragma unroll
        for (int e = 0; e < 8; ++e) wv[e] += c2bias;   // conv2_b on every edge

        // ---- agg += W_tile @ h_jtile via WMMA f32 16x16x4 ----
        const float* hbj = hb + (size_t)jBase * C;
#pragma unroll
        for (int q = 0; q < 4; ++q) {
            v2f afrag;
            afrag.x = wv[q * 2 + 0];
            afrag.y = wv[q * 2 + 1];
            const int krow = 4 * q + 2 * half;   // B-fragment rows for this lane
#pragma unroll
            for (int cb = 0; cb < 4; ++cb) {
                v2f bfrag;
                bfrag.x = hbj[(size_t)(krow + 0) * C + cb * 16 + ncol];
                bfrag.y = hbj[(size_t)(krow + 1) * C + cb * 16 + ncol];
                accs[cb] = __builtin_amdgcn_wmma_f32_16x16x4_f32(
                    false, afrag, false, bfrag, (short)0, accs[cb], false, false);
            }
        }
    }

    // ---- reduce 4 waves' partial agg; final linear + residual ----
    __syncthreads();                 // all waves done using uJbuf as u_j storage
    float* aggP = uJbuf;             // reuse: [NWAVE][TI*C]
#pragma unroll
    for (int cb = 0; cb < 4; ++cb)
#pragma unroll
        for (int r = 0; r < 8; ++r)
            aggP[wave * TI * C + (r + 8 * half) * C + cb * 16 + ncol] = accs[cb][r];
    __syncthreads();

    // sum partials in place: 1024 elements / 128 threads = 8 each
#pragma unroll
    for (int e = 0; e < 8; ++e) {
        const int idx = tid * 8 + e;
        aggP[idx] = aggP[idx] + aggP[TI * C + idx] +
                    aggP[2 * TI * C + idx] + aggP[3 * TI * C + idx];
    }
    __syncthreads();

    // out[iBase+frow, o] = h + elu(agg @ lin_w^T + lin_b)
    const int frow  = tid >> 3;          // 0..15
    const int obase = (tid & 7) * 8;     // 0..56
    float areg[C];
#pragma unroll
    for (int c = 0; c < C; c += 4)
        *(float4*)&areg[c] = *(const float4*)&aggP[frow * C + c];

    const float* hrow = hb + (size_t)(iBase + frow) * C;
    const size_t ooff = out_off(b, iBase + frow);
    float res[8];
#pragma unroll
    for (int oo = 0; oo < 8; ++oo) {
        const int o = obase + oo;
        const float* lw = linw + (size_t)o * C;
        float v = linb[o];
#pragma unroll
        for (int c = 0; c < C; ++c) v = fmaf(areg[c], lw[c], v);
        res[oo] = hrow[o] + eluf(v);
    }
    *(float4*)&out[ooff + obase + 0] = *(float4*)&res[0];
    *(float4*)&out[ooff + obase + 4] = *(float4*)&res[4];
}

// ---------------------------------------------------------------------------
// Host-side launcher
// ---------------------------------------------------------------------------
extern "C" void kernel_launch(void* const* d_in, const int* in_sizes, int n_in,
                              void* d_out, int out_size, void* d_ws, size_t ws_size,
                              hipStream_t stream) {
    (void)in_sizes; (void)n_in; (void)out_size; (void)ws_size;

    const float* xp   = (const float*)d_in[0];   // x_pred  [2,256,64]
    const float* xy   = (const float*)d_in[1];   // x_prey  [2,512,64]
    const float* xo   = (const float*)d_in[2];   // x_obst  [2,256,64]
    const float* linw = (const float*)d_in[3];   // lin_w   [64,64]
    const float* linb = (const float*)d_in[4];   // lin_b   [64]
    const float* w1   = (const float*)d_in[5];   // conv1_w [128,64]
    const float* b1   = (const float*)d_in[6];   // conv1_b [128]
    const float* w2   = (const float*)d_in[7];   // conv2_w [1,128]
    const float* c2b  = (const float*)d_in[8];   // conv2_b [1]

    float* h = (float*)d_ws;                     // [B,N,C]   = 512 KB
    float* u = h + (size_t)BB * N * C;           // [B,N,2C]  = 1 MB
    float* out = (float*)d_out;

    prep_kernel<<<BB * N, 128, 0, stream>>>(xp, xy, xo, w1, h, u);
    edge_agg_kernel<<<BB * (N / TI), 128, 0, stream>>>(h, u, b1, w2, linw, linb, c2b, out);
}